// Tacotron_42296837931488
// MI455X (gfx1250) — compile-verified
//
#include <hip/hip_runtime.h>
#include <hip/hip_bf16.h>
#include <math.h>

// ---------------- sizes ----------------
#define BB    64
#define TX    2048
#define DD    256
#define LL    1024
#define NMELS 80
#define MAXR  20
#define RR    2

typedef float v2f __attribute__((ext_vector_type(2)));
typedef float v8f __attribute__((ext_vector_type(8)));

__device__ __forceinline__ float sigmoidf_(float x) { return 1.0f / (1.0f + expf(-x)); }

// ---------------------------------------------------------------------------
// fp32 WMMA 16x16 tile accumulate:  acc += X[m0:m0+16, :K] * W[n0:n0+16, :K]^T
// Per ISA 7.12.2 (32-bit A 16x4): lanes 0-15 hold K = k0+0 (v0), k0+1 (v1);
// lanes 16-31 hold K = k0+2, k0+3.  B (4x16) mirrors this per-half layout.
// ---------------------------------------------------------------------------
__device__ __forceinline__ v8f wmma_acc_f32(v8f acc,
                                            const float* __restrict__ Xrow, int ldx,
                                            const float* __restrict__ Wrow, int ldw,
                                            int K, int idx, int koff) {
  #pragma unroll 4
  for (int k0 = 0; k0 < K; k0 += 4) {
    const float* xp = Xrow + idx * ldx + k0 + koff;
    const float* wp = Wrow + idx * ldw + k0 + koff;
    v2f a; a.x = xp[0]; a.y = xp[1];
    v2f b; b.x = wp[0]; b.y = wp[1];
    acc = __builtin_amdgcn_wmma_f32_16x16x4_f32(false, a, false, b, (short)0, acc,
                                                false, false);
  }
  return acc;
}

// Generic (optionally dual-input) GEMM:  Out = X1*W1^T [+ X2*W2^T] + bias1 [+ bias2]
// One wave computes one 16x16 output tile. total = Mtiles*Ntiles waves.
__global__ void wmma_gemm_kernel(const float* __restrict__ X1, int ldx1,
                                 const float* __restrict__ W1, int ldw1, int K1,
                                 const float* __restrict__ X2, int ldx2,
                                 const float* __restrict__ W2, int ldw2, int K2,
                                 const float* __restrict__ bias1,
                                 const float* __restrict__ bias2,
                                 float* __restrict__ Out, int ldo,
                                 int Ntiles, int total) {
  int wave = (int)((blockIdx.x * blockDim.x + threadIdx.x) >> 5);
  if (wave >= total) return;                       // wave-uniform: EXEC stays full
  int lane = (int)(threadIdx.x & 31);
  int half = lane >> 4;                            // 0 or 1
  int idx  = lane & 15;
  int m0 = (wave / Ntiles) * 16;
  int n0 = (wave % Ntiles) * 16;
  int koff = 2 * half;

  v8f acc = {0.f, 0.f, 0.f, 0.f, 0.f, 0.f, 0.f, 0.f};
  acc = wmma_acc_f32(acc, X1 + (size_t)m0 * ldx1, ldx1,
                          W1 + (size_t)n0 * ldw1, ldw1, K1, idx, koff);
  if (X2)
    acc = wmma_acc_f32(acc, X2 + (size_t)m0 * ldx2, ldx2,
                            W2 + (size_t)n0 * ldw2, ldw2, K2, idx, koff);

  float bsum = 0.f;
  if (bias1) bsum += bias1[n0 + idx];
  if (bias2) bsum += bias2[n0 + idx];

  // C/D layout: VGPR r holds rows M = r (lanes 0-15) and M = r+8 (lanes 16-31)
  float* outp = Out + (size_t)m0 * ldo + n0;
  #pragma unroll
  for (int r = 0; r < 8; ++r)
    outp[(size_t)(r + 8 * half) * ldo + idx] = acc[r] + bsum;
}

// ---------------- small dense helpers (tiny layers) ----------------
__global__ void fc_relu_kernel(const float* __restrict__ X, int ldx, int K,
                               const float* __restrict__ W,
                               const float* __restrict__ bias,
                               float* __restrict__ Out, int N, int total) {
  int idx = (int)(blockIdx.x * blockDim.x + threadIdx.x);
  if (idx >= total) return;
  int b = idx / N, o = idx % N;
  const float* x = X + (size_t)b * ldx;
  const float* w = W + (size_t)o * K;
  float s = bias[o];
  for (int k = 0; k < K; ++k) s += x[k] * w[k];
  Out[idx] = fmaxf(s, 0.f);
}

__global__ void concat_kernel(const float* __restrict__ A, int na,
                              const float* __restrict__ B, int nb,
                              float* __restrict__ Out, int total) {
  int idx = (int)(blockIdx.x * blockDim.x + threadIdx.x);
  if (idx >= total) return;
  int ld = na + nb;
  int b = idx / ld, k = idx % ld;
  Out[idx] = (k < na) ? A[(size_t)b * na + k] : B[(size_t)b * nb + (k - na)];
}

// ---------------- GRU gate fusion: h' = (1-z)*n + z*h ----------------
__global__ void gru_gate_kernel(const float* __restrict__ gi,
                                const float* __restrict__ gh,
                                const float* __restrict__ hold,
                                float* __restrict__ hnew) {
  int idx = (int)(blockIdx.x * blockDim.x + threadIdx.x);   // 64*256
  int b = idx >> 8, j = idx & 255;
  const float* gib = gi + (size_t)b * (3 * DD);
  const float* ghb = gh + (size_t)b * (3 * DD);
  float r = sigmoidf_(gib[j]        + ghb[j]);
  float z = sigmoidf_(gib[j + DD]   + ghb[j + DD]);
  float n = tanhf    (gib[j + 2*DD] + r * ghb[j + 2*DD]);
  hnew[idx] = (1.f - z) * n + z * hold[idx];
}

// ---------------- LSTM gate fusion (+ residual x) ----------------
__global__ void lstm_gate_kernel(const float* __restrict__ g,
                                 const float* __restrict__ cold,
                                 const float* __restrict__ xin,
                                 float* __restrict__ hnew,
                                 float* __restrict__ cnew,
                                 float* __restrict__ xout) {
  int idx = (int)(blockIdx.x * blockDim.x + threadIdx.x);   // 64*1024
  int b = idx >> 10, j = idx & 1023;
  const float* gb = g + (size_t)b * (4 * LL);
  float i  = sigmoidf_(gb[j]);
  float f  = sigmoidf_(gb[j + LL]);
  float gg = tanhf    (gb[j + 2 * LL]);
  float o  = sigmoidf_(gb[j + 3 * LL]);
  float c  = f * cold[idx] + i * gg;
  float h  = o * tanhf(c);
  cnew[idx] = c;
  hnew[idx] = h;
  xout[idx] = xin[idx] + h;
}

// ---------------------------------------------------------------------------
// Bahdanau attention, one block per batch row, 256 threads.
// Pass A: wave per timestep -> u[t] = v . tanh(enc[t] + qp)   (coalesced 1KB)
// Softmax over t in LDS.  Pass B: thread-per-d context, coalesced over d.
// Writes xcat[b] = [ctx | h_in[b]] directly (the only consumer layout).
// ---------------------------------------------------------------------------
__global__ void attention_kernel(const float* __restrict__ enc,
                                 const float* __restrict__ qp,
                                 const float* __restrict__ v,
                                 const float* __restrict__ h_in,
                                 float* __restrict__ xcat, int T) {
  __shared__ float u[TX];
  __shared__ float qs[DD];
  __shared__ float vs[DD];
  __shared__ float red[8];
  int b = (int)blockIdx.x;
  int tid = (int)threadIdx.x;           // 256 threads, 8 waves
  int wave = tid >> 5, lane = tid & 31;
  qs[tid] = qp[(size_t)b * DD + tid];
  vs[tid] = v[tid];
  __syncthreads();

  const float* encb = enc + (size_t)b * T * DD;

  // ---- scores ----
  for (int t = wave; t < T; t += 8) {
    const float* e = encb + (size_t)t * DD;
    float s = 0.f;
    #pragma unroll
    for (int j = 0; j < 8; ++j) {
      int d = lane + 32 * j;
      s += tanhf(e[d] + qs[d]) * vs[d];
    }
    #pragma unroll
    for (int off = 16; off > 0; off >>= 1) s += __shfl_xor(s, off, 32);
    if (lane == 0) u[t] = s;
  }
  __syncthreads();

  // ---- softmax (max, exp, sum) ----
  float mloc = -3.0e38f;
  for (int t = tid; t < T; t += 256) mloc = fmaxf(mloc, u[t]);
  #pragma unroll
  for (int off = 16; off > 0; off >>= 1) mloc = fmaxf(mloc, __shfl_xor(mloc, off, 32));
  if (lane == 0) red[wave] = mloc;
  __syncthreads();
  float mx = red[0];
  #pragma unroll
  for (int w = 1; w < 8; ++w) mx = fmaxf(mx, red[w]);
  __syncthreads();                       // all reads of red done before reuse

  float sloc = 0.f;
  for (int t = tid; t < T; t += 256) { float e = expf(u[t] - mx); u[t] = e; sloc += e; }
  #pragma unroll
  for (int off = 16; off > 0; off >>= 1) sloc += __shfl_xor(sloc, off, 32);
  if (lane == 0) red[wave] = sloc;
  __syncthreads();                       // also fences the u[] updates
  float tot = 0.f;
  #pragma unroll
  for (int w = 0; w < 8; ++w) tot += red[w];
  float inv = 1.f / tot;

  // ---- context (second pass: L2-resident on MI455X, 134MB < 192MB L2) ----
  int d = tid;
  const float* ep = encb + d;
  float acc = 0.f;
  for (int t = 0; t < T; ++t) {
    if ((t & 31) == 0 && (t + 32) < T)
      __builtin_prefetch(ep + (size_t)(t + 32) * DD, 0, 1);
    acc += u[t] * ep[(size_t)t * DD];
  }
  xcat[(size_t)b * (2 * DD) + d]      = acc * inv;
  xcat[(size_t)b * (2 * DD) + DD + d] = h_in[(size_t)b * DD + d];
}

// ---------------- mel projection: only r<2 of 20 columns survive ----------------
__global__ void melproj_kernel(const float* __restrict__ x,
                               const float* __restrict__ W,
                               float* __restrict__ out) {
  int gw = (int)((blockIdx.x * blockDim.x + threadIdx.x) >> 5);
  if (gw >= BB * NMELS * RR) return;
  int lane = (int)(threadIdx.x & 31);
  int b = gw / (NMELS * RR), n = gw % (NMELS * RR);
  int m = n >> 1, r = n & 1;
  const float* w  = W + (size_t)(m * MAXR + r) * LL;
  const float* xb = x + (size_t)b * LL;
  float s = 0.f;
  for (int k = lane; k < LL; k += 32) s += xb[k] * w[k];
  #pragma unroll
  for (int off = 16; off > 0; off >>= 1) s += __shfl_xor(s, off, 32);
  if (lane == 0) out[(size_t)b * (NMELS * RR) + n] = s;
}

// ---------------------------------------------------------------------------
extern "C" void kernel_launch(void* const* d_in, const int* in_sizes, int n_in,
                              void* d_out, int out_size, void* d_ws, size_t ws_size,
                              hipStream_t stream) {
  const float* enc       = (const float*)d_in[0];
  const float* prenet_in = (const float*)d_in[2];
  const float* h_text    = (const float*)d_in[3];
  const float* h_attn    = (const float*)d_in[5];
  const float* h_r1      = (const float*)d_in[6];
  const float* h_r2      = (const float*)d_in[7];
  const float* c_r1      = (const float*)d_in[8];
  const float* c_r2      = (const float*)d_in[9];
  const float* cv_text   = (const float*)d_in[10];
  const float* cv        = (const float*)d_in[12];
  const float* fc1_w = (const float*)d_in[13]; const float* fc1_b = (const float*)d_in[14];
  const float* fc2_w = (const float*)d_in[15]; const float* fc2_b = (const float*)d_in[16];
  const float* tW    = (const float*)d_in[17]; const float* tv    = (const float*)d_in[18];
  const float* t_wih = (const float*)d_in[21]; const float* t_whh = (const float*)d_in[22];
  const float* t_bih = (const float*)d_in[23]; const float* t_bhh = (const float*)d_in[24];
  const float* a_wih = (const float*)d_in[29]; const float* a_whh = (const float*)d_in[30];
  const float* a_bih = (const float*)d_in[31]; const float* a_bhh = (const float*)d_in[32];
  const float* ri_w  = (const float*)d_in[33]; const float* ri_b  = (const float*)d_in[34];
  const float* l1_wih = (const float*)d_in[35]; const float* l1_whh = (const float*)d_in[36];
  const float* l1_bih = (const float*)d_in[37]; const float* l1_bhh = (const float*)d_in[38];
  const float* l2_wih = (const float*)d_in[39]; const float* l2_whh = (const float*)d_in[40];
  const float* l2_bih = (const float*)d_in[41]; const float* l2_bhh = (const float*)d_in[42];
  const float* mp_w   = (const float*)d_in[43];
  float* out = (float*)d_out;

  // ---- workspace layout (floats) ----
  float* ws = (float*)d_ws;
  float* p1      = ws;                    // [64,256]
  float* p       = p1      + 16384;       // [64,128]
  float* x_text  = p       + 8192;        // [64,384]
  float* x_attn  = x_text  + 24576;       // [64,384]
  float* gi_text = x_attn  + 24576;       // [64,768]
  float* gh_text = gi_text + 49152;       // [64,768]
  float* gi_attn = gh_text + 49152;       // [64,768]
  float* gh_attn = gi_attn + 49152;       // [64,768]
  float* h_text2 = gh_attn + 49152;       // [64,256]
  float* h_attn2 = h_text2 + 16384;       // [64,256]
  float* qp_text = h_attn2 + 16384;       // [64,256]
  float* xcat    = qp_text + 16384;       // [64,512]
  float* xri     = xcat    + 32768;       // [64,1024]
  float* g1      = xri     + 65536;       // [64,4096]
  float* hr1n    = g1      + 262144;      // [64,1024]
  float* cr1n    = hr1n    + 65536;       // [64,1024]
  float* x2      = cr1n    + 65536;       // [64,1024]
  float* g2      = x2      + 65536;       // [64,4096]
  float* hr2n    = g2      + 262144;      // [64,1024]
  float* cr2n    = hr2n    + 65536;       // unused downstream, kept for clarity
  float* x3      = cr2n    + 65536;       // [64,1024]

  const dim3 blk(256);

  // 1) PreNet
  fc_relu_kernel<<<dim3(64), blk, 0, stream>>>(prenet_in, NMELS, NMELS, fc1_w, fc1_b,
                                               p1, 256, 64 * 256);
  fc_relu_kernel<<<dim3(32), blk, 0, stream>>>(p1, 256, 256, fc2_w, fc2_b,
                                               p, 128, 64 * 128);
  concat_kernel<<<dim3(96), blk, 0, stream>>>(cv_text, DD, p, 128, x_text, 64 * 384);
  concat_kernel<<<dim3(96), blk, 0, stream>>>(cv,      DD, p, 128, x_attn, 64 * 384);

  // 2) GRU pre-activations (style branch is dead code -> skipped entirely)
  // gi = x @ wih^T + bih  (K=384, N=768);  gh = h @ whh^T + bhh (K=256, N=768)
  wmma_gemm_kernel<<<dim3(24), blk, 0, stream>>>(x_text, 384, t_wih, 384, 384,
                                                 nullptr, 0, nullptr, 0, 0,
                                                 t_bih, nullptr, gi_text, 768, 48, 192);
  wmma_gemm_kernel<<<dim3(24), blk, 0, stream>>>(h_text, 256, t_whh, 256, 256,
                                                 nullptr, 0, nullptr, 0, 0,
                                                 t_bhh, nullptr, gh_text, 768, 48, 192);
  wmma_gemm_kernel<<<dim3(24), blk, 0, stream>>>(x_attn, 384, a_wih, 384, 384,
                                                 nullptr, 0, nullptr, 0, 0,
                                                 a_bih, nullptr, gi_attn, 768, 48, 192);
  wmma_gemm_kernel<<<dim3(24), blk, 0, stream>>>(h_attn, 256, a_whh, 256, 256,
                                                 nullptr, 0, nullptr, 0, 0,
                                                 a_bhh, nullptr, gh_attn, 768, 48, 192);
  gru_gate_kernel<<<dim3(64), blk, 0, stream>>>(gi_text, gh_text, h_text, h_text2);
  gru_gate_kernel<<<dim3(64), blk, 0, stream>>>(gi_attn, gh_attn, h_attn, h_attn2);

  // 3) qp = h_text' @ tW^T  (K=256, N=256), then text attention -> xcat=[ctx|h_attn']
  wmma_gemm_kernel<<<dim3(8), blk, 0, stream>>>(h_text2, 256, tW, 256, 256,
                                                nullptr, 0, nullptr, 0, 0,
                                                nullptr, nullptr, qp_text, 256, 16, 64);
  attention_kernel<<<dim3(BB), blk, 0, stream>>>(enc, qp_text, tv, h_attn2, xcat, TX);

  // 4) x = xcat @ ri_w^T + ri_b  (K=512, N=1024)
  wmma_gemm_kernel<<<dim3(32), blk, 0, stream>>>(xcat, 512, ri_w, 512, 512,
                                                 nullptr, 0, nullptr, 0, 0,
                                                 ri_b, nullptr, xri, 1024, 64, 256);

  // 5) LSTM 1: g = x@wih^T + h@whh^T + biases (dual GEMM, K=1024 each, N=4096)
  wmma_gemm_kernel<<<dim3(128), blk, 0, stream>>>(xri, 1024, l1_wih, 1024, 1024,
                                                  h_r1, 1024, l1_whh, 1024, 1024,
                                                  l1_bih, l1_bhh, g1, 4096, 256, 1024);
  lstm_gate_kernel<<<dim3(256), blk, 0, stream>>>(g1, c_r1, xri, hr1n, cr1n, x2);

  // 6) LSTM 2
  wmma_gemm_kernel<<<dim3(128), blk, 0, stream>>>(x2, 1024, l2_wih, 1024, 1024,
                                                  h_r2, 1024, l2_whh, 1024, 1024,
                                                  l2_bih, l2_bhh, g2, 4096, 256, 1024);
  lstm_gate_kernel<<<dim3(256), blk, 0, stream>>>(g2, c_r2, x2, hr2n, cr2n, x3);

  // 7) mel projection: only the r<2 slice of the 20-wide reshape is needed
  melproj_kernel<<<dim3(1280), blk, 0, stream>>>(x3, mp_w, out);

  (void)in_sizes; (void)n_in; (void)out_size; (void)ws_size;
}